// GravityMCDecoder_5162550690703
// MI455X (gfx1250) — compile-verified
//
#include <hip/hip_runtime.h>

#define EPS      0.01f
#define N_NODES  12288
#define D_PLUS_1 65
#define N_EDGES  262144

typedef float v2f __attribute__((ext_vector_type(2)));
typedef float v8f __attribute__((ext_vector_type(8)));

// ---------------------------------------------------------------------------
// Kernel 1: per-node squared norm of r = z[:, :64].  One wave32 per node:
// lane k loads elements k and k+32, squares, butterfly-reduces across the wave.
// 12288 nodes, 8 waves per 256-thread block -> 1536 blocks. ~3 MB streamed.
// ---------------------------------------------------------------------------
__global__ __launch_bounds__(256) void gravity_norms_kernel(
    const float* __restrict__ z, float* __restrict__ norm) {
  const int wave = (int)((blockIdx.x * blockDim.x + threadIdx.x) >> 5);
  const int lane = (int)(threadIdx.x & 31);
  if (wave >= N_NODES) return;
  const float* row = z + (size_t)wave * D_PLUS_1;
  const float a = row[lane];
  const float b = row[lane + 32];
  float s = a * a + b * b;
#pragma unroll
  for (int off = 16; off >= 1; off >>= 1)
    s += __shfl_xor(s, off, 32);
  if (lane == 0) norm[wave] = s;
}

// ---------------------------------------------------------------------------
// Kernel 2: one wave32 per 16 edges.  The 16 dot products r_u . r_v are the
// diagonal of U(16x64) x V^T(64x16), accumulated with 16x
// V_WMMA_F32_16X16X4_F32 on the matrix pipe.
//
// A-matrix (16x4 f32) layout: lane = M (mod 16); lanes 0-15 carry K=0,1 in
// VGPR0/1, lanes 16-31 carry K=2,3.  B mirrors with N = lane&15.  So each
// lane reads only ONE u-row and ONE v-row -> every row is fetched exactly
// once per wave (512 B/edge, the minimum; all hits in the 192 MB L2 since
// z is only 3.2 MB).
//
// Diagonal (i,i) of the f32 16x16 C tile sits in VGPR (i&7) at lane i (i<8)
// or lane 16+i (i>=8); extracted via a small LDS bounce.
// ---------------------------------------------------------------------------
__global__ __launch_bounds__(128) void gravity_edge_kernel(
    const float* __restrict__ z, const float* __restrict__ lscale,
    const int* __restrict__ eli, const float* __restrict__ norm,
    float* __restrict__ out) {
  __shared__ float sh[4][256];  // 4 waves/block x (32 lanes x 8 VGPRs)

  const int lane  = (int)(threadIdx.x & 31);
  const int w     = (int)(threadIdx.x >> 5);
  const int tile  = (int)(blockIdx.x * 4 + w);  // 16 edges per tile
  const int eIdx  = tile * 16 + (lane & 15);

  const int u = eli[eIdx];
  const int v = eli[N_EDGES + eIdx];
  const float* __restrict__ urow = z + (size_t)u * D_PLUS_1;
  const float* __restrict__ vrow = z + (size_t)v * D_PLUS_1;

  const int koff = (lane >> 4) << 1;  // 0 for lanes 0-15, 2 for lanes 16-31

  v8f c = {0.f, 0.f, 0.f, 0.f, 0.f, 0.f, 0.f, 0.f};
#pragma unroll
  for (int kk = 0; kk < 64; kk += 4) {
    v2f a, b;
    a.x = urow[kk + koff];
    a.y = urow[kk + koff + 1];
    b.x = vrow[kk + koff];
    b.y = vrow[kk + koff + 1];
    // D = A x B + C  (accumulate into c)
    c = __builtin_amdgcn_wmma_f32_16x16x4_f32(
        /*neg_a=*/false, a, /*neg_b=*/false, b,
        /*c_mod=*/(short)0, c, /*reuse_a=*/false, /*reuse_b=*/false);
  }

  // Spill the C tile so each lane<16 can pick up its diagonal element.
#pragma unroll
  for (int j = 0; j < 8; ++j) sh[w][lane * 8 + j] = c[j];
  __syncthreads();

  if (lane < 16) {
    const int srcLane = (lane < 8) ? lane : lane + 16;
    const float dot = sh[w][srcLane * 8 + (lane & 7)];

    const float nu = norm[u];
    const float nv = norm[v];
    const float mu = urow[64];  // m = z[:, -1]
    const float mv = vrow[64];
    const float ls = lscale[0];

    const float r2 = nu + nv - 2.0f * dot;
    const float lr = __logf(r2 + EPS);
    const float logit_uv = mv - ls * lr;  // adj[u,v]
    const float logit_vu = mu - ls * lr;  // adj[v,u]
    const float suv = 1.0f / (1.0f + __expf(-logit_uv));
    const float svu = 1.0f / (1.0f + __expf(-logit_vu));

    float4 p;
    p.x = (1.0f - suv) * (1.0f - svu);  // p_nb
    p.y = suv * (1.0f - svu);           // p_pu
    p.z = suv * svu;                    // p_pb
    p.w = (1.0f - suv) * svu;           // p_nu
    *(float4*)(out + (size_t)eIdx * 4) = p;
  }
}

extern "C" void kernel_launch(void* const* d_in, const int* in_sizes, int n_in,
                              void* d_out, int out_size, void* d_ws, size_t ws_size,
                              hipStream_t stream) {
  const float* z      = (const float*)d_in[0];
  const float* lscale = (const float*)d_in[1];
  const int*   eli    = (const int*)d_in[2];   // int32 (JAX default, no x64)
  float*       out    = (float*)d_out;
  float*       norm   = (float*)d_ws;          // 12288 floats of scratch

  (void)in_sizes; (void)n_in; (void)out_size; (void)ws_size;

  // 12288 nodes / (8 waves per block) = 1536 blocks
  gravity_norms_kernel<<<N_NODES / 8, 256, 0, stream>>>(z, norm);
  // 262144 edges / (16 per wave * 4 waves per block) = 4096 blocks
  gravity_edge_kernel<<<N_EDGES / 64, 128, 0, stream>>>(z, lscale, eli, norm, out);
}